// DfOpOneStep_55765855371911
// MI455X (gfx1250) — compile-verified
//
#include <hip/hip_runtime.h>
#include <stdint.h>

// DfOpOneStep: bandwidth-bound streaming op.
//   spec_frame: (B,1,1,481,2) f32   -> float2 idx: b*481 + f
//   coefs_frame:(B,5,1,96,2)  f32   -> float2 idx: b*480 + k*96 + f
//   spec_buf:   (B,1,5,96,2)  f32   -> float2 idx: b*480 + k*96 + f
// out = [enhanced (B*481*2 f32)] ++ [new_spec_buf (B*480*2 f32)]

typedef __attribute__((ext_vector_type(2))) float f2;

#define BATCH 32768
#define NF    96
#define ORD   5
#define FF    481
#define HI    (FF - NF)   // 385 pass-through complex bins per batch

// ---------------------------------------------------------------------------
// Kernel 1: ring-buffer shift + 5-tap complex MAC for the low 96 bins.
// One thread per (b, f) complex bin. All accesses are coalesced B64 with
// non-temporal hints (working set 630 MB >> 192 MB L2; each byte used once).
// ---------------------------------------------------------------------------
__global__ __launch_bounds__(256) void df_mac_kernel(
    const f2* __restrict__ spec_frame,   // B*481
    const f2* __restrict__ coefs,        // B*480
    const f2* __restrict__ spec_buf,     // B*480
    f2* __restrict__ enhanced,           // B*481
    f2* __restrict__ new_buf)            // B*480
{
    unsigned idx = blockIdx.x * 256u + threadIdx.x;   // [0, BATCH*NF)
    unsigned b = idx / NF;
    unsigned f = idx - b * NF;

    const f2* sb = spec_buf + (size_t)b * (ORD * NF) + f;
    const f2* cf = coefs    + (size_t)b * (ORD * NF) + f;

    // Shifted buffer: new_buf[k] = spec_buf[k+1] (k<4), new_buf[4] = new frame
    f2 s[ORD];
#pragma unroll
    for (int k = 0; k < ORD - 1; ++k)
        s[k] = __builtin_nontemporal_load(sb + (k + 1) * NF);
    s[ORD - 1] = __builtin_nontemporal_load(spec_frame + (size_t)b * FF + f);

    float re = 0.f, im = 0.f;
#pragma unroll
    for (int k = 0; k < ORD; ++k) {
        f2 c = __builtin_nontemporal_load(cf + k * NF);
        re = fmaf(s[k].x,  c.x, re);
        re = fmaf(-s[k].y, c.y, re);
        im = fmaf(s[k].x,  c.y, im);
        im = fmaf(s[k].y,  c.x, im);
    }

    f2 o; o.x = re; o.y = im;
    __builtin_nontemporal_store(o, enhanced + (size_t)b * FF + f);

    f2* nb = new_buf + (size_t)b * (ORD * NF) + f;
#pragma unroll
    for (int k = 0; k < ORD; ++k)
        __builtin_nontemporal_store(s[k], nb + k * NF);
}

// ---------------------------------------------------------------------------
// Kernel 2: pass-through copy of bins [96,481) via the CDNA5 async
// Global->LDS->Global path (ASYNCcnt-tracked, no VGPR staging).
// Each lane moves 4 float2 per block: 4 async loads into LDS, one
// s_wait_asynccnt, 4 async stores back out, one final wait.
// Grid is exact: BATCH*HI float2 = 12,615,680 = 12320 blocks * 1024.
// ---------------------------------------------------------------------------
__global__ __launch_bounds__(256) void df_copy_kernel(
    const float* __restrict__ src,   // spec_frame base
    float* __restrict__ dst)         // enhanced base
{
    __shared__ f2 stage[4][256];
    const unsigned t = threadIdx.x;

    unsigned lds[4];
    unsigned off[4];
#pragma unroll
    for (int j = 0; j < 4; ++j) {
        // Flat LDS pointers keep the LDS byte offset in the low 32 bits.
        lds[j] = (unsigned)(uintptr_t)&stage[j][t];
        unsigned i = blockIdx.x * 1024u + j * 256u + t;   // float2 index
        unsigned b = i / HI;
        unsigned r = i - b * HI;
        off[j] = (b * (unsigned)FF + (unsigned)NF + r) * 8u;  // byte offset
    }

#pragma unroll
    for (int j = 0; j < 4; ++j)
        asm volatile("global_load_async_to_lds_b64 %0, %1, %2"
                     :: "v"(lds[j]), "v"(off[j]), "s"(src) : "memory");

    asm volatile("s_wait_asynccnt 0" ::: "memory");

#pragma unroll
    for (int j = 0; j < 4; ++j)
        asm volatile("global_store_async_from_lds_b64 %0, %1, %2"
                     :: "v"(off[j]), "v"(lds[j]), "s"(dst) : "memory");

    asm volatile("s_wait_asynccnt 0" ::: "memory");
}

extern "C" void kernel_launch(void* const* d_in, const int* in_sizes, int n_in,
                              void* d_out, int out_size, void* d_ws, size_t ws_size,
                              hipStream_t stream) {
    const float* spec_frame = (const float*)d_in[0];
    const float* coefs      = (const float*)d_in[1];
    const float* spec_buf   = (const float*)d_in[2];

    float* out      = (float*)d_out;
    float* enhanced = out;                           // BATCH*FF*2 floats
    float* new_buf  = out + (size_t)BATCH * FF * 2;  // BATCH*ORD*NF*2 floats

    df_mac_kernel<<<BATCH * NF / 256, 256, 0, stream>>>(
        (const f2*)spec_frame, (const f2*)coefs, (const f2*)spec_buf,
        (f2*)enhanced, (f2*)new_buf);

    df_copy_kernel<<<BATCH * HI / 1024, 256, 0, stream>>>(spec_frame, enhanced);
}